// VBLDS_62053687492989
// MI455X (gfx1250) — compile-verified
//
#include <hip/hip_runtime.h>
#include <math.h>

typedef __attribute__((ext_vector_type(2))) float v2f;
typedef __attribute__((ext_vector_type(8))) float v8f;

#define LANE ((int)threadIdx.x & 31)

constexpr int N  = 32768;
constexpr int DZ = 16;
constexpr int DX = 32;
constexpr float TWO_PI_F = 6.28318530717958647692f;

// d_out flat layout (return order): ell, muf(dz,N), Vf(dz,dz,N), mus(dz,N), Vs(dz,dz,N), V12(dz,dz,N-1)
constexpr int OFF_MUF = 1;
constexpr int OFF_VF  = OFF_MUF + DZ * N;
constexpr int OFF_MUS = OFF_VF + DZ * DZ * N;
constexpr int OFF_VS  = OFF_MUS + DZ * N;
constexpr int OFF_V12 = OFF_VS + DZ * DZ * N;

// workspace (float offsets)
constexpr int WS_XX  = 0;     // 32x32
constexpr int WS_ZZ  = 1024;  // 16x16
constexpr int WS_Z11 = 1280;
constexpr int WS_Z12 = 1536;
constexpr int WS_Z22 = 1792;
constexpr int WS_ZX  = 2048;  // 16x32 row-major

// ---------------------------------------------------------------------------
// CDNA5 async Global->LDS copy (ASYNCcnt-tracked), used for scan double
// buffering so the L2 load latency of the next step's operands is hidden
// behind the current step's WMMA chain.
// ---------------------------------------------------------------------------
__device__ __forceinline__ unsigned lds_off_of(const volatile void* p) {
  typedef __attribute__((address_space(3))) const volatile void lds_t;
  return (unsigned)(unsigned long long)(lds_t*)p;
}

__device__ __forceinline__ void async_gl2lds_b32(unsigned ldsoff, unsigned goff,
                                                 const float* base) {
  // GVS mode: mem_addr = SGPR64 + VGPR32 ; LDS dest offset in VDST VGPR.
  asm volatile("global_load_async_to_lds_b32 %0, %1, %2 offset:0"
               :
               : "v"(ldsoff), "v"(goff), "s"(base)
               : "memory");
}

__device__ __forceinline__ void wait_async0() {
  asm volatile("s_wait_asynccnt 0x0" ::: "memory");
}

// ---------------------------------------------------------------------------
// Wave-level 16x16 f32 matmul via V_WMMA_F32_16X16X4_F32 (4 chained K=4 ops).
// A frag (16x4, 32-bit): lane holds M=lane&15, VGPR v holds K = v + 2*(lane>=16)
// B frag (4x16): VGPR v holds row K = v + 2*(lane>=16), col N = lane&15
// C/D frag (16x16): VGPR v = row v + 8*(lane>=16), col = lane&15
// ---------------------------------------------------------------------------
__device__ __forceinline__ v8f wmma_mm16(const float* A, int lda, bool ta,
                                         const float* B, int ldb, bool tb, v8f acc) {
  const int lo = LANE & 15, hi = LANE >> 4;
#pragma unroll
  for (int kc = 0; kc < 4; ++kc) {
    v2f a, b;
#pragma unroll
    for (int t = 0; t < 2; ++t) {
      const int k = 4 * kc + t + 2 * hi;
      a[t] = ta ? A[k * lda + lo] : A[lo * lda + k];
      b[t] = tb ? B[lo * ldb + k] : B[k * ldb + lo];
    }
    acc = __builtin_amdgcn_wmma_f32_16x16x4_f32(false, a, false, b, (short)0, acc,
                                                false, false);
  }
  return acc;
}

__device__ __forceinline__ v8f frag_load(const float* C, int ldc) {
  const int lo = LANE & 15, hi = LANE >> 4;
  v8f c;
#pragma unroll
  for (int v = 0; v < 8; ++v) c[v] = C[(v + 8 * hi) * ldc + lo];
  return c;
}

__device__ __forceinline__ void frag_store(float* D, int ldd, v8f c) {
  const int lo = LANE & 15, hi = LANE >> 4;
#pragma unroll
  for (int v = 0; v < 8; ++v) D[(v + 8 * hi) * ldd + lo] = c[v];
}

// D = op(A)*op(B) (+ C if C != nullptr); all LDS, barriers included.
__device__ __forceinline__ void mm16(float* D, int ldd, const float* A, int lda,
                                     bool ta, const float* B, int ldb, bool tb,
                                     const float* C, int ldc) {
  __syncthreads();
  v8f acc = {};
  if (C) acc = frag_load(C, ldc);
  acc = wmma_mm16(A, lda, ta, B, ldb, tb, acc);
  frag_store(D, ldd, acc);
  __syncthreads();
}

// Gauss-Jordan inverse NNxNN in LDS (lane l owns row l). No pivoting: all
// matrices here are SPD / near-identity. aug must hold NN*LD floats, LD>=2*NN.
template <int NN, int LD>
__device__ __forceinline__ void gj_inv(const float* M, int ldm, float* out, int ldo,
                                       float* aug) {
  const int l = LANE;
  __syncthreads();
  if (l < NN) {
    for (int j = 0; j < NN; ++j) {
      aug[l * LD + j] = M[l * ldm + j];
      aug[l * LD + NN + j] = (l == j) ? 1.0f : 0.0f;
    }
  }
  __syncthreads();
  for (int k = 0; k < NN; ++k) {
    const float piv = aug[k * LD + k];
    const float pinv = 1.0f / piv;
    __syncthreads();
    if (l == k) {
      for (int j = 0; j < 2 * NN; ++j) aug[k * LD + j] *= pinv;
    }
    __syncthreads();
    if (l < NN && l != k) {
      const float f = aug[l * LD + k];
      for (int j = 0; j < 2 * NN; ++j) aug[l * LD + j] -= f * aug[k * LD + j];
    }
    __syncthreads();
  }
  if (l < NN)
    for (int j = 0; j < NN; ++j) out[l * ldo + j] = aug[l * LD + NN + j];
  __syncthreads();
}

// out16[16] = M(16x16) @ v16
__device__ __forceinline__ void mv16(float* out, const float* M, const float* v) {
  const int l = LANE;
  __syncthreads();
  float s = 0.0f;
  if (l < 16) {
    for (int j = 0; j < 16; ++j) s += M[l * 16 + j] * v[j];
  }
  __syncthreads();
  if (l < 16) out[l] = s;
  __syncthreads();
}

// out = I - V @ inv(I + S@V) @ S  (all 16x16 ld16)
__device__ __forceinline__ void woodbury16(float* out, const float* V, const float* S,
                                           float* t1, float* t2, float* ti,
                                           float* aug) {
  const int l = LANE;
  mm16(t1, 16, S, 16, false, V, 16, false, nullptr, 0);  // t1 = S@V
  if (l < 16) t1[l * 16 + l] += 1.0f;
  __syncthreads();
  gj_inv<16, 33>(t1, 16, ti, 16, aug);                   // ti = inv(I+SV)
  mm16(t2, 16, V, 16, false, ti, 16, false, nullptr, 0); // t2 = V@ti
  mm16(t1, 16, t2, 16, false, S, 16, false, nullptr, 0); // t1 = t2@S
  for (int e = l; e < 256; e += 32) {
    const int r = e >> 4, c = e & 15;
    out[e] = ((r == c) ? 1.0f : 0.0f) - t1[e];
  }
  __syncthreads();
}

// ---------------------------------------------------------------------------
// Parallel xx = x @ x^T reduction (only data-parallel heavy piece).
// ---------------------------------------------------------------------------
__global__ void __launch_bounds__(256) k_xx_zero(float* ws) {
  const int t = blockIdx.x * blockDim.x + threadIdx.x;
  if (t < 1024) ws[WS_XX + t] = 0.0f;
}

__global__ void __launch_bounds__(256) k_xx(const float* __restrict__ x,
                                            float* __restrict__ ws) {
  __shared__ float sx[32 * 64];
  const int t = threadIdx.x;
  const int n0b = blockIdx.x * 256;  // 128 blocks * 256 = N
  float acc[4] = {0.f, 0.f, 0.f, 0.f};
  for (int tile = 0; tile < 4; ++tile) {
    const int n0 = n0b + tile * 64;
    __syncthreads();
#pragma unroll
    for (int k = 0; k < 8; ++k) {
      const int e = t + k * 256;
      const int row = e >> 6, col = e & 63;
      sx[e] = x[row * N + n0 + col];
    }
    __syncthreads();
#pragma unroll
    for (int k = 0; k < 4; ++k) {
      const int p = t + k * 256;
      const int i = p >> 5, j = p & 31;
      float s = 0.f;
      for (int c = 0; c < 64; ++c) s += sx[i * 64 + c] * sx[j * 64 + c];
      acc[k] += s;
    }
  }
#pragma unroll
  for (int k = 0; k < 4; ++k) atomicAdd(&ws[WS_XX + t + k * 256], acc[k]);
}

// ---------------------------------------------------------------------------
// Kalman filter: sequential forward scan, one wave. Writes muf, Vf.
// x columns are async-prefetched into a double buffer one step ahead.
// ---------------------------------------------------------------------------
__global__ void __launch_bounds__(32) k_filter(
    const float* __restrict__ x, const float* __restrict__ A,
    const float* __restrict__ C, const float* __restrict__ Q,
    const float* __restrict__ R, const float* __restrict__ m0,
    const float* __restrict__ P0, const float* __restrict__ Saqa,
    const float* __restrict__ Scrc, float* __restrict__ out) {
  __shared__ float sA[256], sCc[512], sQ[256], sR[1024], sScrc[256], sSaqa[256];
  __shared__ float sm[16], sP[256];
  __shared__ float sT1[256], sT2[256], sTi[256], sL[256], sLP[256];
  __shared__ float sCLP[512];    // 32x16
  __shared__ float sShat[1056];  // 32x32, ld 33
  __shared__ float sInvS[1056];  // 32x32, ld 33
  __shared__ float sM1[528];     // 16x32, ld 33  (LP @ C^T)
  __shared__ float sK[528];      // 16x32, ld 33
  __shared__ float sAug[2080];   // 32x65 GJ scratch (also reused 16x33)
  __shared__ float sV[256], sG[256], sAG[256], sAGV[256];
  __shared__ float sMu[16], sLm[16], sGm[16], sXh[32];
  __shared__ float sXb[2][32];   // async double buffer for x[:, n]

  const int l = LANE;
  for (int e = l; e < 256; e += 32) {
    sA[e] = A[e]; sQ[e] = Q[e]; sScrc[e] = Scrc[e]; sSaqa[e] = Saqa[e]; sP[e] = P0[e];
  }
  for (int e = l; e < 512; e += 32) sCc[e] = C[e];
  for (int e = l; e < 1024; e += 32) sR[e] = R[e];
  if (l < 16) sm[l] = m0[l];
  // prefetch x[:,0]
  async_gl2lds_b32(lds_off_of(&sXb[0][l]), (unsigned)(l * N) * 4u, x);
  __syncthreads();

  float* muf = out + OFF_MUF;
  float* Vf = out + OFF_VF;

  for (int n = 0; n < N; ++n) {
    const int buf = n & 1;
    wait_async0();
    __syncthreads();
    if (n + 1 < N)  // prefetch x[:, n+1] into other buffer
      async_gl2lds_b32(lds_off_of(&sXb[buf ^ 1][l]),
                       (unsigned)(l * N + n + 1) * 4u, x);
    // L = I - P inv(I + Scrc P) Scrc ; Lm ; LP
    woodbury16(sL, sP, sScrc, sT1, sT2, sTi, sAug);
    mv16(sLm, sL, sm);
    mm16(sLP, 16, sL, 16, false, sP, 16, false, nullptr, 0);
    // residual = x_n - C@Lm (lane l = obs row l)
    {
      float s = 0.f;
      for (int j = 0; j < 16; ++j) s += sCc[l * 16 + j] * sLm[j];
      sXh[l] = sXb[buf][l] - s;
    }
    __syncthreads();
    // CLP = C @ LP (32x16)
    mm16(sCLP, 16, sCc, 16, false, sLP, 16, false, nullptr, 0);
    mm16(sCLP + 256, 16, sCc + 256, 16, false, sLP, 16, false, nullptr, 0);
    // Shat = CLP @ C^T + R (32x32, ld33)
    for (int bi = 0; bi < 2; ++bi)
      for (int bj = 0; bj < 2; ++bj)
        mm16(sShat + bi * 16 * 33 + bj * 16, 33, sCLP + bi * 256, 16, false,
             sCc + bj * 256, 16, true, nullptr, 0);
    for (int e = l; e < 1024; e += 32) {
      const int r = e >> 5, c = e & 31;
      sShat[r * 33 + c] += sR[e];
    }
    __syncthreads();
    gj_inv<32, 65>(sShat, 33, sInvS, 33, sAug);
    // M1 = LP @ C^T (16x32)
    mm16(sM1, 33, sLP, 16, false, sCc, 16, true, nullptr, 0);
    mm16(sM1 + 16, 33, sLP, 16, false, sCc + 256, 16, true, nullptr, 0);
    // K = M1 @ invS (16x32)
    for (int bj = 0; bj < 2; ++bj) {
      mm16(sK + bj * 16, 33, sM1, 33, false, sInvS + bj * 16, 33, false, nullptr, 0);
      mm16(sK + bj * 16, 33, sM1 + 16, 33, false, sInvS + 16 * 33 + bj * 16, 33,
           false, sK + bj * 16, 33);
    }
    // mu = Lm + K @ residual
    if (l < 16) {
      float s = sLm[l];
      for (int j = 0; j < 32; ++j) s += sK[l * 33 + j] * sXh[j];
      sMu[l] = s;
    }
    __syncthreads();
    // KC = K@C ; V = sym((I-KC) @ LP)
    mm16(sT1, 16, sK, 33, false, sCc, 16, false, nullptr, 0);
    mm16(sT1, 16, sK + 16, 33, false, sCc + 256, 16, false, sT1, 16);
    for (int e = l; e < 256; e += 32) {
      const int r = e >> 4, c = e & 15;
      sT2[e] = ((r == c) ? 1.0f : 0.0f) - sT1[e];
    }
    __syncthreads();
    mm16(sT1, 16, sT2, 16, false, sLP, 16, false, nullptr, 0);
    for (int e = l; e < 256; e += 32) {
      const int r = e >> 4, c = e & 15;
      sV[e] = 0.5f * (sT1[e] + sT1[c * 16 + r]);
    }
    __syncthreads();
    // outputs
    if (l < 16) muf[l * N + n] = sMu[l];
    for (int e = l; e < 256; e += 32) Vf[e * N + n] = sV[e];
    // predict: G = woodbury(V,Saqa); m' = A G mu; P' = sym(A G V A^T) + Q
    woodbury16(sG, sV, sSaqa, sT1, sT2, sTi, sAug);
    mv16(sGm, sG, sMu);
    mv16(sm, sA, sGm);
    mm16(sAG, 16, sA, 16, false, sG, 16, false, nullptr, 0);
    mm16(sAGV, 16, sAG, 16, false, sV, 16, false, nullptr, 0);
    mm16(sT1, 16, sAGV, 16, false, sA, 16, true, nullptr, 0);
    for (int e = l; e < 256; e += 32) {
      const int r = e >> 4, c = e & 15;
      sP[e] = 0.5f * (sT1[e] + sT1[c * 16 + r]) + sQ[e];
    }
    __syncthreads();
  }
}

// ---------------------------------------------------------------------------
// RTS smoother: sequential reverse scan + moment accumulation. One wave.
// Filter values (muf, Vf) for the next step are async-prefetched from L2
// into a double-buffered LDS tile while the current step computes.
// ---------------------------------------------------------------------------
__global__ void __launch_bounds__(32) k_smooth(
    const float* __restrict__ x, const float* __restrict__ A,
    const float* __restrict__ Q, const float* __restrict__ Saqa,
    float* __restrict__ out, float* __restrict__ ws) {
  __shared__ float sA[256], sQ[256], sSaqa[256];
  __shared__ float sMuP[16], sVP[256];        // carry (mus[n+1], Vs[n+1])
  __shared__ float sVfn[2][256], sMufn[2][16];  // async double buffers
  __shared__ float sG[256], sGV[256], sT1[256], sT2[256], sTi[256];
  __shared__ float sAug[528];
  __shared__ float sP[256], sJ[256], sVnew[256];
  __shared__ float sGm[16], sM[16], sDm[16], sMu[16];

  const int l = LANE;
  const int lo = l & 15, hi = l >> 4;
  for (int e = l; e < 256; e += 32) { sA[e] = A[e]; sQ[e] = Q[e]; sSaqa[e] = Saqa[e]; }
  // init carry from last filter step
  for (int e = l; e < 256; e += 32) sVP[e] = out[OFF_VF + e * N + (N - 1)];
  if (l < 16) sMuP[l] = out[OFF_MUF + l * N + (N - 1)];
  // prefetch filter values for n = N-2 into buffer 0
  {
    const int n = N - 2;
    for (int e = l; e < 256; e += 32)
      async_gl2lds_b32(lds_off_of(&sVfn[0][e]), (unsigned)(e * N + n) * 4u,
                       out + OFF_VF);
    if (l < 16)
      async_gl2lds_b32(lds_off_of(&sMufn[0][l]), (unsigned)(l * N + n) * 4u,
                       out + OFF_MUF);
  }
  __syncthreads();
  // tail outputs: mus[N-1]=muf[N-1], Vs[N-1]=Vf[N-1]
  if (l < 16) out[OFF_MUS + l * N + (N - 1)] = sMuP[l];
  for (int e = l; e < 256; e += 32) out[OFF_VS + e * N + (N - 1)] = sVP[e];

  // moment accumulators in WMMA C-fragment layout: elem (v+8*hi, lo)
  float azz[8], az11[8], az12[8], az22[8], azx[16];
#pragma unroll
  for (int v = 0; v < 8; ++v) {
    const int r = v + 8 * hi;
    const float mm = sMuP[r] * sMuP[lo] + sVP[r * 16 + lo];
    azz[v] = mm;   // n = N-1 term
    az22[v] = mm;  // n = N-1 term
    az11[v] = 0.f;
    az12[v] = 0.f;
  }
  {
    const float xv = x[l * N + (N - 1)];
#pragma unroll
    for (int i = 0; i < 16; ++i) azx[i] = sMuP[i] * xv;
  }
  __syncthreads();

  for (int n = N - 2; n >= 0; --n) {
    const int buf = (N - 2 - n) & 1;
    wait_async0();  // current buffer complete
    __syncthreads();
    if (n > 0) {  // prefetch n-1 into the other buffer
      const int nb = buf ^ 1, np = n - 1;
      for (int e = l; e < 256; e += 32)
        async_gl2lds_b32(lds_off_of(&sVfn[nb][e]), (unsigned)(e * N + np) * 4u,
                         out + OFF_VF);
      if (l < 16)
        async_gl2lds_b32(lds_off_of(&sMufn[nb][l]), (unsigned)(l * N + np) * 4u,
                         out + OFF_MUF);
    }
    const float* Vfn = sVfn[buf];
    const float* Mufn = sMufn[buf];
    // G = woodbury(Vf_n, Saqa); GV = G@Vf_n
    woodbury16(sG, Vfn, sSaqa, sT1, sT2, sTi, sAug);
    mm16(sGV, 16, sG, 16, false, Vfn, 16, false, nullptr, 0);
    mv16(sGm, sG, Mufn);
    mv16(sM, sA, sGm);  // predicted mean m
    // P = sym(A @ GV @ A^T) + Q
    mm16(sT1, 16, sA, 16, false, sGV, 16, false, nullptr, 0);
    mm16(sT2, 16, sT1, 16, false, sA, 16, true, nullptr, 0);
    for (int e = l; e < 256; e += 32) {
      const int r = e >> 4, c = e & 15;
      sP[e] = 0.5f * (sT2[e] + sT2[c * 16 + r]) + sQ[e];
    }
    __syncthreads();
    // J = GV @ A^T @ inv(P)
    mm16(sT1, 16, sGV, 16, false, sA, 16, true, nullptr, 0);
    gj_inv<16, 33>(sP, 16, sTi, 16, sAug);
    mm16(sJ, 16, sT1, 16, false, sTi, 16, false, nullptr, 0);
    // mu = G muf_n + J (mu_plus - m)
    if (l < 16) sDm[l] = sMuP[l] - sM[l];
    __syncthreads();
    if (l < 16) {
      float s = sGm[l];
      for (int j = 0; j < 16; ++j) s += sJ[l * 16 + j] * sDm[j];
      sMu[l] = s;
    }
    __syncthreads();
    // V = sym(GV + J (V_plus - P) J^T)
    for (int e = l; e < 256; e += 32) sT2[e] = sVP[e] - sP[e];
    __syncthreads();
    mm16(sT1, 16, sJ, 16, false, sT2, 16, false, nullptr, 0);
    mm16(sT2, 16, sT1, 16, false, sJ, 16, true, nullptr, 0);
    for (int e = l; e < 256; e += 32) {
      const int r = e >> 4, c = e & 15;
      sVnew[e] = sGV[e] + 0.5f * (sT2[e] + sT2[c * 16 + r]);
    }
    __syncthreads();
    // V12 = J @ V_plus (direct WMMA fragment, written straight to global)
    v8f f12 = {};
    f12 = wmma_mm16(sJ, 16, false, sVP, 16, false, f12);
    // outputs
    if (l < 16) out[OFF_MUS + l * N + n] = sMu[l];
    for (int e = l; e < 256; e += 32) out[OFF_VS + e * N + n] = sVnew[e];
#pragma unroll
    for (int v = 0; v < 8; ++v) {
      const int r = v + 8 * hi;
      out[OFF_V12 + (r * 16 + lo) * (N - 1) + n] = f12[v];
      const float mm = sMu[r] * sMu[lo] + sVnew[r * 16 + lo];
      azz[v] += mm;
      az11[v] += mm;                          // n in [0, N-2]
      az12[v] += sMu[r] * sMuP[lo] + f12[v];  // outer(mus[n], mus[n+1]) + V12
      if (n > 0) az22[v] += mm;               // n in [1, N-1]
    }
    {
      const float xv = x[l * N + n];
#pragma unroll
      for (int i = 0; i < 16; ++i) azx[i] += sMu[i] * xv;
    }
    __syncthreads();
    // carry update
    if (l < 16) sMuP[l] = sMu[l];
    for (int e = l; e < 256; e += 32) sVP[e] = sVnew[e];
    __syncthreads();
  }

  // dump accumulators
#pragma unroll
  for (int v = 0; v < 8; ++v) {
    const int e = (v + 8 * hi) * 16 + lo;
    ws[WS_ZZ + e] = azz[v];
    ws[WS_Z11 + e] = az11[v];
    ws[WS_Z12 + e] = az12[v];
    ws[WS_Z22 + e] = az22[v];
  }
#pragma unroll
  for (int i = 0; i < 16; ++i) ws[WS_ZX + i * 32 + l] = azx[i];
}

// ---------------------------------------------------------------------------
// Finalize: M-step + ELBO pieces -> out[0]. One wave.
// ---------------------------------------------------------------------------
__global__ void __launch_bounds__(32) k_final(const float* __restrict__ ws,
                                              float* __restrict__ out) {
  __shared__ float sZZ[256], sZ11[256], sZ12[256], sZ22[256], sZX[512];
  __shared__ float sSig[256], sAn[256], sCn[512], sCZ[512];
  __shared__ float sT1[256], sTi[256], sAug[528], sP0[256];
  __shared__ float sRed[32];
  const int l = LANE;
  for (int e = l; e < 256; e += 32) {
    sZZ[e] = ws[WS_ZZ + e]; sZ11[e] = ws[WS_Z11 + e];
    sZ12[e] = ws[WS_Z12 + e]; sZ22[e] = ws[WS_Z22 + e];
  }
  for (int e = l; e < 512; e += 32) sZX[e] = ws[WS_ZX + e];
  __syncthreads();

  float ell_acc = 0.0f;  // valid on lane 0

  // ---- learn_AQ: Sigma_A = inv(z1z1 + 1e-6 I); A_new = z1z2^T Sigma_A ----
  for (int e = l; e < 256; e += 32) {
    const int r = e >> 4, c = e & 15;
    sT1[e] = sZ11[e] + ((r == c) ? 1e-6f : 0.0f);
  }
  __syncthreads();
  gj_inv<16, 33>(sT1, 16, sSig, 16, sAug);
  mm16(sAn, 16, sZ12, 16, true, sSig, 16, false, nullptr, 0);
  mm16(sT1, 16, sAn, 16, false, sZ11, 16, false, nullptr, 0);  // A_new @ z1z1
  {
    float pz = 0.0f;
    if (l < 16) {
      float dot = 0.f, q2 = 0.f;
      for (int j = 0; j < 16; ++j) {
        dot += sAn[l * 16 + j] * sZ12[j * 16 + l];
        q2 += sT1[l * 16 + j] * sAn[l * 16 + j];
      }
      const float c_hat = 1e-9f + 0.5f * (float)(N - 1);
      const float d_hat = 1e-9f + 0.5f * (sZ22[l * 16 + l] - dot);
      const float Qd = d_hat / c_hat;
      const float quad = sZ22[l * 16 + l] - 2.0f * dot + q2;
      pz = -0.5f * (float)(N - 1) * logf(TWO_PI_F * Qd) - 0.5f * quad / Qd;
    }
    sRed[l] = pz;
    __syncthreads();
    if (l == 0)
      for (int i = 0; i < 32; ++i) ell_acc += sRed[i];
    __syncthreads();
  }

  // ---- learn_CR: Sigma_C = inv(zz + 1e-6 I); C_new = zx^T Sigma_C ----
  for (int e = l; e < 256; e += 32) {
    const int r = e >> 4, c = e & 15;
    sT1[e] = sZZ[e] + ((r == c) ? 1e-6f : 0.0f);
  }
  __syncthreads();
  gj_inv<16, 33>(sT1, 16, sSig, 16, sAug);
  mm16(sCn, 16, sZX, 32, true, sSig, 16, false, nullptr, 0);             // rows 0..15
  mm16(sCn + 256, 16, sZX + 16, 32, true, sSig, 16, false, nullptr, 0);  // rows 16..31
  mm16(sCZ, 16, sCn, 16, false, sZZ, 16, false, nullptr, 0);
  mm16(sCZ + 256, 16, sCn + 256, 16, false, sZZ, 16, false, nullptr, 0);
  {
    float dot = 0.f, q2 = 0.f;
    for (int j = 0; j < 16; ++j) {
      dot += sCn[l * 16 + j] * sZX[j * 32 + l];
      q2 += sCZ[l * 16 + j] * sCn[l * 16 + j];
    }
    const float xxii = ws[WS_XX + l * 32 + l];
    const float a_hat = 1e-9f + 0.5f * (float)N;
    const float b_hat = 1e-9f + 0.5f * (xxii - dot);
    const float Rd = b_hat / a_hat;
    const float quad = xxii - 2.0f * dot + q2;
    sRed[l] = -0.5f * (float)N * logf(TWO_PI_F * Rd) - 0.5f * quad / Rd;
    __syncthreads();
    if (l == 0)
      for (int i = 0; i < 32; ++i) ell_acc += sRed[i];
    __syncthreads();
  }

  // ---- ell_z0: P0_new = sym(Vs[0]); quad0 == P0_new => trace term = dz ----
  for (int e = l; e < 256; e += 32) {
    const int r = e >> 4, c = e & 15;
    const float a = out[OFF_VS + e * N];
    const float b = out[OFF_VS + (c * 16 + r) * N];
    sP0[e] = 0.5f * (a + b);
  }
  __syncthreads();
  if (l < 16)
    for (int j = 0; j < 16; ++j) sAug[l * 33 + j] = sP0[l * 16 + j];
  __syncthreads();
  float ldet = 0.0f;  // lane 0
  for (int k = 0; k < 16; ++k) {
    const float piv = sAug[k * 33 + k];
    if (l == 0) ldet += logf(piv);
    const float pinv = 1.0f / piv;
    __syncthreads();
    if (l < 16 && l > k) {
      const float f = sAug[l * 33 + k] * pinv;
      for (int j = k; j < 16; ++j) sAug[l * 33 + j] -= f * sAug[k * 33 + j];
    }
    __syncthreads();
  }
  if (l == 0) {
    const float ell_z0 = -0.5f * (16.0f * logf(TWO_PI_F) + ldet) - 8.0f;
    out[0] = ell_acc + ell_z0;
  }
}

// ---------------------------------------------------------------------------
extern "C" void kernel_launch(void* const* d_in, const int* in_sizes, int n_in,
                              void* d_out, int out_size, void* d_ws, size_t ws_size,
                              hipStream_t stream) {
  const float* x = (const float*)d_in[0];
  const float* A = (const float*)d_in[1];
  const float* C = (const float*)d_in[2];
  const float* Q = (const float*)d_in[3];
  const float* R = (const float*)d_in[4];
  const float* m0 = (const float*)d_in[5];
  const float* P0 = (const float*)d_in[6];
  const float* Saqa = (const float*)d_in[7];
  const float* Scrc = (const float*)d_in[8];
  float* out = (float*)d_out;
  float* ws = (float*)d_ws;

  hipLaunchKernelGGL(k_xx_zero, dim3(4), dim3(256), 0, stream, ws);
  hipLaunchKernelGGL(k_xx, dim3(128), dim3(256), 0, stream, x, ws);
  hipLaunchKernelGGL(k_filter, dim3(1), dim3(32), 0, stream, x, A, C, Q, R, m0, P0,
                     Saqa, Scrc, out);
  hipLaunchKernelGGL(k_smooth, dim3(1), dim3(32), 0, stream, x, A, Q, Saqa, out, ws);
  hipLaunchKernelGGL(k_final, dim3(1), dim3(32), 0, stream, ws, out);
}